// StructuredStateSpace_80736795230386
// MI455X (gfx1250) — compile-verified
//
#include <hip/hip_runtime.h>
#include <stdint.h>

#define D_STATE 512
#define SEQ     4096
#define BATCH   8
#define CH      32                         // timesteps per TDM tile
#define CHBYTES (CH * D_STATE * 4)         // 64 KB per tile

typedef uint32_t u32;
typedef uint64_t u64;
typedef __attribute__((ext_vector_type(4))) u32 v4u;
typedef __attribute__((ext_vector_type(8))) int v8i;
typedef __attribute__((ext_vector_type(4))) int v4i;

// Build a 2-D TDM descriptor (D#) and issue TENSOR_LOAD_TO_LDS:
//   tile = CH rows x 512 cols of f32, row stride 512, into LDS at lds_off.
static __device__ __forceinline__ void tdm_load_chunk(u32 lds_off, u64 gaddr) {
  v4u g0;
  g0.x = 1u;                                              // count=1, is_restore=0, gather=0
  g0.y = lds_off;                                         // LDS byte address
  g0.z = (u32)(gaddr & 0xffffffffu);                      // global_addr[31:0]
  g0.w = (u32)((gaddr >> 32) & 0x01ffffffu) | (2u << 30); // global_addr[56:32] | type=2
  v8i g1;
  g1[0] = (2 << 16);                          // workgroup_mask=0, data_size=2 (4B)
  g1[1] = (int)((u32)D_STATE << 16);          // tensor_dim0[15:0]=512 in bits[31:16]
  g1[2] = (int)((u32)(SEQ & 0xffff) << 16);   // tensor_dim0 hi=0 | tensor_dim1 lo16=4096
  g1[3] = (int)((u32)D_STATE << 16);          // tensor_dim1 hi=0 | tile_dim0=512
  g1[4] = CH;                                 // tile_dim1=CH, tile_dim2=0
  g1[5] = D_STATE;                            // tensor_dim0_stride = 512
  g1[6] = 0;
  g1[7] = 0;
  v4i gz4 = (v4i){0, 0, 0, 0};
  v8i gz8 = (v8i){0, 0, 0, 0, 0, 0, 0, 0};
  // 6-arg form (clang-23 / therock-10.0 headers): groups 2/3 disabled (zeros)
  __builtin_amdgcn_tensor_load_to_lds(g0, g1, gz4, gz4, gz8, 0);
}

__global__ __launch_bounds__(32)
void ssm_scan_kernel(const float* __restrict__ x,
                     const float* __restrict__ vol,
                     const float* __restrict__ log_lambda_real,
                     const float* __restrict__ log_b,
                     const float* __restrict__ c,
                     const float* __restrict__ log_step,
                     const float* __restrict__ vgate,
                     const float* __restrict__ ln_w,
                     const float* __restrict__ ln_b,
                     float* __restrict__ s_out)
{
  extern __shared__ float xbuf[];            // 2 * CH * 512 floats (double buffer)
  const int b    = blockIdx.x;
  const int lane = threadIdx.x;

  // Per-lane parameters for the 16 owned states: n = lane*16 + j
  float a_d[16], b_d[16], cv[16], gt[16], w_[16], bb[16], h[16];
#pragma unroll
  for (int j = 0; j < 16; ++j) {
    const int n    = lane * 16 + j;
    const float lam = -expf(log_lambda_real[n]);
    const float st  =  expf(log_step[n]);
    const float bv  =  expf(log_b[n]);
    const float z   = st * lam;
    const float ad  = (2.0f + z) / (2.0f - z);   // bilinear (Tustin)
    a_d[j] = ad;
    b_d[j] = st * (1.0f + ad) * bv * 0.5f;
    cv[j]  = c[n];
    gt[j]  = 1.0f / (1.0f + expf(-vgate[n]));    // sigmoid(volatility_gate)
    w_[j]  = ln_w[n];
    bb[j]  = ln_b[n];
    h[j]   = 0.0f;
  }

  const float* xb = x   + (size_t)b * SEQ * D_STATE;
  const float* vb = vol + (size_t)b * SEQ;
  float*       sb = s_out + (size_t)b * SEQ;
  const int nchunks = SEQ / CH;

  tdm_load_chunk(0u, (u64)(uintptr_t)xb);      // prime the pipeline

  for (int ck = 0; ck < nchunks; ++ck) {
    if (ck + 1 < nchunks) {
      tdm_load_chunk((u32)(((ck + 1) & 1) * CHBYTES),
                     (u64)(uintptr_t)(xb + (size_t)(ck + 1) * CH * D_STATE));
      __builtin_amdgcn_s_wait_tensorcnt(1);    // current tile done, next in flight
    } else {
      __builtin_amdgcn_s_wait_tensorcnt(0);
    }

    const float vv = vb[ck * CH + lane];       // one coalesced load, broadcast below
    const float* xs = &xbuf[(ck & 1) * (CH * D_STATE)];
    float s_keep = 0.0f;

#pragma unroll 4
    for (int t = 0; t < CH; ++t) {
      const float vt = __shfl(vv, t, 32);      // v_t scalar for this step

      // diagonal SSM update (independent FMAs)
#pragma unroll
      for (int j = 0; j < 16; ++j) {
        const float xv = xs[t * D_STATE + lane * 16 + j];
        h[j] = fmaf(a_d[j], h[j], b_d[j] * xv);
      }

      // wave-wide sum / sum-of-squares over the 512 states
      float s0=0,s1=0,s2=0,s3=0, q0=0,q1=0,q2=0,q3=0;
#pragma unroll
      for (int j = 0; j < 16; j += 4) {
        s0 += h[j];   s1 += h[j+1]; s2 += h[j+2]; s3 += h[j+3];
        q0 = fmaf(h[j],  h[j],  q0); q1 = fmaf(h[j+1],h[j+1],q1);
        q2 = fmaf(h[j+2],h[j+2],q2); q3 = fmaf(h[j+3],h[j+3],q3);
      }
      float sum = (s0 + s1) + (s2 + s3);
      float sq  = (q0 + q1) + (q2 + q3);
#pragma unroll
      for (int m = 1; m < 32; m <<= 1) {
        sum += __shfl_xor(sum, m, 32);
        sq  += __shfl_xor(sq,  m, 32);
      }
      const float mu  = sum * (1.0f / 512.0f);
      const float var = fmaf(sq, 1.0f / 512.0f, -mu * mu);  // E[h^2] - mu^2
      const float inv = __builtin_amdgcn_rsqf(var + 1e-5f);

      // layernorm, volatility gate, and c-dot partial (off the carry chain)
      float p = 0.0f;
#pragma unroll
      for (int j = 0; j < 16; ++j) {
        const float hn = fmaf((h[j] - mu) * inv, w_[j], bb[j]);
        const float g  = fmaf(gt[j], vt, 1.0f);
        h[j] = hn * __builtin_amdgcn_rcpf(g);
        p = fmaf(h[j], cv[j], p);
      }
#pragma unroll
      for (int m = 1; m < 32; m <<= 1) p += __shfl_xor(p, m, 32);
      if (lane == t) s_keep = p;               // stage for coalesced store
    }
    sb[ck * CH + lane] = s_keep;               // 128B coalesced store per chunk
  }
}

// out[b,t,d] = (al + (1-al)*exp(log_d)) * x + (1-al) * s[b,t]   (bandwidth-bound)
__global__ __launch_bounds__(256)
void ssm_out_kernel(const float4* __restrict__ x4,
                    const float* __restrict__ s,
                    const float* __restrict__ log_d,
                    const float* __restrict__ alpha,
                    float4* __restrict__ o4)
{
  const int i  = blockIdx.x * blockDim.x + threadIdx.x;   // exact grid, no tail
  const float al = 1.0f / (1.0f + expf(-alpha[0]));
  const float dd = expf(log_d[0]);
  const float cx = al + (1.0f - al) * dd;
  const float sv = (1.0f - al) * s[i >> 7];               // 128 float4 per (b,t) row
  float4 xv = x4[i];
  float4 r;
  r.x = fmaf(cx, xv.x, sv);
  r.y = fmaf(cx, xv.y, sv);
  r.z = fmaf(cx, xv.z, sv);
  r.w = fmaf(cx, xv.w, sv);
  o4[i] = r;
}

extern "C" void kernel_launch(void* const* d_in, const int* in_sizes, int n_in,
                              void* d_out, int out_size, void* d_ws, size_t ws_size,
                              hipStream_t stream) {
  const float* x       = (const float*)d_in[0];
  const float* vol     = (const float*)d_in[1];
  const float* llr     = (const float*)d_in[2];
  const float* logb    = (const float*)d_in[3];
  const float* c       = (const float*)d_in[4];
  const float* logd    = (const float*)d_in[5];
  const float* logstep = (const float*)d_in[6];
  const float* vg      = (const float*)d_in[7];
  const float* alpha   = (const float*)d_in[8];
  const float* lnw     = (const float*)d_in[9];
  const float* lnb     = (const float*)d_in[10];

  float* s = (float*)d_ws;                    // BATCH*SEQ floats = 128 KB scratch

  const size_t shbytes = (size_t)2 * CH * D_STATE * sizeof(float);  // 128 KB LDS
  hipLaunchKernelGGL(ssm_scan_kernel, dim3(BATCH), dim3(32), shbytes, stream,
                     x, vol, llr, logb, c, logstep, vg, lnw, lnb, s);

  const int total4 = BATCH * SEQ * D_STATE / 4;            // 4,194,304
  hipLaunchKernelGGL(ssm_out_kernel, dim3(total4 / 256), dim3(256), 0, stream,
                     (const float4*)x, s, logd, alpha, (float4*)d_out);
}